// GeneralAffinityCalculator_55697135894755
// MI455X (gfx1250) — compile-verified
//
#include <hip/hip_runtime.h>
#include <hip/hip_bf16.h>

// Problem constants (from reference)
#define LATENT_DIM 128
#define KQ_DIM      32
#define BATCH        4
#define NNODES   16384
#define KEDGE       64
#define MTOT   (BATCH * NNODES)          // 65536 rows
#define NEDGE  ((size_t)BATCH * NNODES * KEDGE)  // 4,194,304 edges

typedef __attribute__((ext_vector_type(2))) float v2f;
typedef __attribute__((ext_vector_type(8))) float v8f;

// ---------------------------------------------------------------------------
// Phase 1: ks/qs GEMM via V_WMMA_F32_16X16X4_F32 (full fp32 precision).
// One wave -> 16 rows of output, 4 accumulator tiles:
//   (Wk, d0=0), (Wk, d0=16), (Wq, d0=0), (Wq, d0=16)
// A fragment (16x4 f32): lane<16 holds {k0, k0+1} of row M=lane;
//                        lane>=16 holds {k0+2, k0+3} of row M=lane-16.
// B fragment (4x16 f32): lane<16 holds {k0, k0+1} of col N=lane;
//                        lane>=16 holds {k0+2, k0+3} of col N=lane-16.
//   B[k][n] = W[n][k]  (ks = features @ W^T), W rows are 128 contiguous floats
//   so each fragment half is a natural v2f load from W.
// C/D (16x16 f32, 8 VGPRs): vgpr r, lane<16 -> (M=r, N=lane);
//                           lane>=16 -> (M=r+8, N=lane-16).
// ---------------------------------------------------------------------------
__global__ __launch_bounds__(256) void kq_gemm_wmma_kernel(
    const float* __restrict__ features,  // [MTOT, 128]
    const float* __restrict__ Wk,        // [32, 128]
    const float* __restrict__ bk,        // [32]
    const float* __restrict__ Wq,        // [32, 128]
    const float* __restrict__ bq,        // [32]
    float* __restrict__ ks,              // [MTOT, 32]
    float* __restrict__ qs)              // [MTOT, 32]
{
  const int lane = threadIdx.x & 31;
  const int wave = threadIdx.x >> 5;
  const int m0   = (blockIdx.x * 8 + wave) * 16;   // 16 rows per wave
  const int half = lane >> 4;                      // 0: lanes 0-15, 1: 16-31
  const int l15  = lane & 15;

  // Bias-initialized accumulators (column d = d0 + l15 for every row).
  v8f ck0, ck1, cq0, cq1;
  {
    const float bk0 = bk[l15], bk1 = bk[l15 + 16];
    const float bq0 = bq[l15], bq1 = bq[l15 + 16];
#pragma unroll
    for (int r = 0; r < 8; ++r) { ck0[r] = bk0; ck1[r] = bk1; cq0[r] = bq0; cq1[r] = bq1; }
  }

  // Per-lane base pointers.
  const float* arow = features + (size_t)(m0 + l15) * LATENT_DIM + half * 2;
  const float* wk0  = Wk + (size_t)(l15)      * LATENT_DIM + half * 2;
  const float* wk1  = Wk + (size_t)(l15 + 16) * LATENT_DIM + half * 2;
  const float* wq0  = Wq + (size_t)(l15)      * LATENT_DIM + half * 2;
  const float* wq1  = Wq + (size_t)(l15 + 16) * LATENT_DIM + half * 2;

#pragma unroll
  for (int k0 = 0; k0 < LATENT_DIM; k0 += 4) {
    const v2f a   = *(const v2f*)(arow + k0);
    const v2f bK0 = *(const v2f*)(wk0 + k0);
    const v2f bK1 = *(const v2f*)(wk1 + k0);
    const v2f bQ0 = *(const v2f*)(wq0 + k0);
    const v2f bQ1 = *(const v2f*)(wq1 + k0);
    ck0 = __builtin_amdgcn_wmma_f32_16x16x4_f32(false, a, false, bK0, (short)0, ck0, false, false);
    ck1 = __builtin_amdgcn_wmma_f32_16x16x4_f32(false, a, false, bK1, (short)0, ck1, false, false);
    cq0 = __builtin_amdgcn_wmma_f32_16x16x4_f32(false, a, false, bQ0, (short)0, cq0, false, false);
    cq1 = __builtin_amdgcn_wmma_f32_16x16x4_f32(false, a, false, bQ1, (short)0, cq1, false, false);
  }

  // Store: vgpr r maps to row m0 + r + 8*half, column l15 (+0 or +16).
  float* ksb = ks + ((size_t)(m0 + 8 * half)) * KQ_DIM + l15;
  float* qsb = qs + ((size_t)(m0 + 8 * half)) * KQ_DIM + l15;
#pragma unroll
  for (int r = 0; r < 8; ++r) {
    ksb[(size_t)r * KQ_DIM + 0]  = ck0[r];
    ksb[(size_t)r * KQ_DIM + 16] = ck1[r];
    qsb[(size_t)r * KQ_DIM + 0]  = cq0[r];
    qsb[(size_t)r * KQ_DIM + 16] = cq1[r];
  }
}

// ---------------------------------------------------------------------------
// Phase 2: per-edge gather + 32-dim dot. One thread per edge.
// Each gathered row is 128 contiguous bytes (one cache line); ks/qs total
// 16.8 MB -> resident in the 192 MB L2, so this runs at L2 gather bandwidth.
// ---------------------------------------------------------------------------
__global__ __launch_bounds__(256) void affinity_kernel(
    const int*   __restrict__ idx_x,   // indices[1]: [B, N*K]
    const int*   __restrict__ idx_y,   // indices[2]: [B, N*K]
    const float* __restrict__ ks,      // [B*N, 32]
    const float* __restrict__ qs,      // [B*N, 32]
    float* __restrict__ out)           // [B, N, K]
{
  const size_t e = (size_t)blockIdx.x * blockDim.x + threadIdx.x;  // < NEDGE
  const int b  = (int)(e >> 20);            // N*K = 2^20
  const int xi = idx_x[e];
  const int yi = idx_y[e];

  const float4* kp = (const float4*)(ks + (((size_t)b << 14) + (size_t)xi) * KQ_DIM);
  const float4* qp = (const float4*)(qs + (((size_t)b << 14) + (size_t)yi) * KQ_DIM);

  float acc = 0.0f;
#pragma unroll
  for (int j = 0; j < 8; ++j) {
    const float4 kv = kp[j];
    const float4 qv = qp[j];
    acc = fmaf(kv.x, qv.x, acc);
    acc = fmaf(kv.y, qv.y, acc);
    acc = fmaf(kv.z, qv.z, acc);
    acc = fmaf(kv.w, qv.w, acc);
  }
  out[e] = acc * 0.17677669529663687f;  // 32^-0.5
}

extern "C" void kernel_launch(void* const* d_in, const int* in_sizes, int n_in,
                              void* d_out, int out_size, void* d_ws, size_t ws_size,
                              hipStream_t stream) {
  // setup_inputs() order: indices, img, features, Wk, bk, Wq, bq
  const int*   indices  = (const int*)d_in[0];     // [3, B, N, K] int32
  const float* features = (const float*)d_in[2];   // [B, N, 128]
  const float* Wk       = (const float*)d_in[3];   // [32, 128]
  const float* bk       = (const float*)d_in[4];   // [32]
  const float* Wq       = (const float*)d_in[5];   // [32, 128]
  const float* bq       = (const float*)d_in[6];   // [32]
  float*       out      = (float*)d_out;           // [B, N, K]

  // Workspace: ks then qs, each [65536, 32] f32 (8 MB each).
  float* ks = (float*)d_ws;
  float* qs = ks + (size_t)MTOT * KQ_DIM;

  // Phase 1: 512 blocks x 8 waves x 16 rows = 65536 rows, all waves full.
  kq_gemm_wmma_kernel<<<MTOT / (8 * 16), 256, 0, stream>>>(
      features, Wk, bk, Wq, bq, ks, qs);

  // Phase 2: one thread per edge.
  const int* idx_x = indices + 1 * NEDGE;  // indices[1]
  const int* idx_y = indices + 2 * NEDGE;  // indices[2]
  affinity_kernel<<<(int)(NEDGE / 256), 256, 0, stream>>>(idx_x, idx_y, ks, qs, out);
}